// SparseGraphLearn_40604620816384
// MI455X (gfx1250) — compile-verified
//
#include <hip/hip_runtime.h>

typedef float v2f __attribute__((ext_vector_type(2)));
typedef float v4f __attribute__((ext_vector_type(4)));
typedef float v8f __attribute__((ext_vector_type(8)));

#define N_NODES 16384
#define DIN     512
#define DOUT    256
#define N_EDGES 524288

// ---------------------------------------------------------------------------
// Zero per-segment max (uint bits) and sum buffers.
// ---------------------------------------------------------------------------
__global__ __launch_bounds__(256) void zero_ms_kernel(unsigned* __restrict__ mbits,
                                                      float* __restrict__ s) {
    int i = blockIdx.x * 256 + threadIdx.x;
    if (i < N_NODES) {
        mbits[i] = 0u;   // bit pattern of +0.0f; valid identity since ev >= 0
        s[i]     = 0.0f;
    }
}

// ---------------------------------------------------------------------------
// Fill the 1.07 GB sgraph region with zeros using B128 non-temporal stores
// (dominant cost of the whole problem; keep it off the L2 that holds h).
// ---------------------------------------------------------------------------
__global__ __launch_bounds__(256) void zero_fill_kernel(v4f* __restrict__ p,
                                                        long long n4) {
    long long stride = (long long)gridDim.x * 256;
    v4f z = {0.f, 0.f, 0.f, 0.f};
    for (long long i = (long long)blockIdx.x * 256 + threadIdx.x; i < n4; i += stride) {
        __builtin_nontemporal_store(z, p + i);
    }
}

// ---------------------------------------------------------------------------
// GEMM: h[N_NODES, DOUT] = x[N_NODES, DIN] * w[DIN, DOUT]  (exact fp32 WMMA)
// One wave computes a 64x16 tile (4 M-subtiles sharing each B fragment).
// Tasks: (N/64) * (DOUT/16) = 4096 waves; 4 waves/block -> 1024 blocks.
// A 16x4 f32 frag: lanes 0-15 row=lane K={0,1}; lanes 16-31 row=lane-16 K={2,3}.
// B 4x16 f32 frag: lanes 0-15 col=lane K={0,1}; lanes 16-31 col=lane-16 K={2,3}.
// D 16x16 f32: vgpr r holds row r (lanes 0-15) / row r+8 (lanes 16-31).
// ---------------------------------------------------------------------------
__global__ __launch_bounds__(128) void gemm_wmma_f32_kernel(const float* __restrict__ x,
                                                            const float* __restrict__ w,
                                                            float* __restrict__ h) {
    const int lane = threadIdx.x & 31;
    const int wave = threadIdx.x >> 5;
    const int task = blockIdx.x * 4 + wave;      // 0..4095
    const int row64 = task >> 4;                 // 0..255
    const int colT  = task & 15;                 // 0..15
    const int rowBase = row64 * 64;
    const int n  = lane & 15;
    const int hi = lane >> 4;                    // 0: K={0,1}, 1: K={2,3}
    const int col = colT * 16 + n;

    v8f c0 = {}, c1 = {}, c2 = {}, c3 = {};

    const float* xrow = x + (size_t)(rowBase + n) * DIN + 2 * hi;
    const float* wp   = w + col;

    #pragma unroll 2
    for (int k = 0; k < DIN; k += 4) {
        v2f a0 = *(const v2f*)(xrow + k);
        v2f a1 = *(const v2f*)(xrow + 16 * DIN + k);
        v2f a2 = *(const v2f*)(xrow + 32 * DIN + k);
        v2f a3 = *(const v2f*)(xrow + 48 * DIN + k);
        v2f b;
        b.x = wp[(size_t)(k + 2 * hi)     * DOUT];
        b.y = wp[(size_t)(k + 2 * hi + 1) * DOUT];
        c0 = __builtin_amdgcn_wmma_f32_16x16x4_f32(false, a0, false, b, (short)0, c0, false, false);
        c1 = __builtin_amdgcn_wmma_f32_16x16x4_f32(false, a1, false, b, (short)0, c1, false, false);
        c2 = __builtin_amdgcn_wmma_f32_16x16x4_f32(false, a2, false, b, (short)0, c2, false, false);
        c3 = __builtin_amdgcn_wmma_f32_16x16x4_f32(false, a3, false, b, (short)0, c3, false, false);
    }

    // Store D tiles: lane row offset 8*hi, vgpr r -> row r.
    float* hp = h + (size_t)(rowBase + 8 * hi) * DOUT + col;
    #pragma unroll
    for (int r = 0; r < 8; ++r) {
        hp[(size_t)(r)      * DOUT] = c0[r];
        hp[(size_t)(16 + r) * DOUT] = c1[r];
        hp[(size_t)(32 + r) * DOUT] = c2[r];
        hp[(size_t)(48 + r) * DOUT] = c3[r];
    }
}

// ---------------------------------------------------------------------------
// Edge scoring: one wave per edge. Each lane handles 8 contiguous channels
// (2x float4), wave-wide shfl_xor reduction, relu, atomic segment-max on
// uint bit pattern (valid ordering for non-negative floats).
// ---------------------------------------------------------------------------
__global__ __launch_bounds__(256) void edge_score_kernel(const float* __restrict__ h,
                                                         const float* __restrict__ a,
                                                         const int* __restrict__ edge,
                                                         float* __restrict__ ev,
                                                         unsigned* __restrict__ mbits) {
    const int lane = threadIdx.x & 31;
    const int e = (blockIdx.x * 256 + threadIdx.x) >> 5;
    if (e >= N_EDGES) return;
    const int src = edge[e];
    const int dst = edge[N_EDGES + e];

    const v4f* hs = (const v4f*)(h + (size_t)src * DOUT) + lane * 2;
    const v4f* hd = (const v4f*)(h + (size_t)dst * DOUT) + lane * 2;
    const v4f* av = (const v4f*)a + lane * 2;

    float acc = 0.f;
    #pragma unroll
    for (int j = 0; j < 2; ++j) {
        v4f s4 = hs[j], d4 = hd[j], a4 = av[j];
        acc += fabsf(s4.x - d4.x) * a4.x;
        acc += fabsf(s4.y - d4.y) * a4.y;
        acc += fabsf(s4.z - d4.z) * a4.z;
        acc += fabsf(s4.w - d4.w) * a4.w;
    }
    #pragma unroll
    for (int off = 16; off > 0; off >>= 1)
        acc += __shfl_xor(acc, off, 32);

    if (lane == 0) {
        float v = fmaxf(acc, 0.f);
        ev[e] = v;
        atomicMax(mbits + dst, __float_as_uint(v));
    }
}

// ---------------------------------------------------------------------------
// ex = exp(ev - m[dst]); segment-sum via atomicAdd.
// ---------------------------------------------------------------------------
__global__ __launch_bounds__(256) void edge_exp_kernel(const int* __restrict__ edge,
                                                       const float* __restrict__ ev,
                                                       const unsigned* __restrict__ mbits,
                                                       float* __restrict__ ex,
                                                       float* __restrict__ s) {
    int e = blockIdx.x * 256 + threadIdx.x;
    if (e >= N_EDGES) return;
    int dst = edge[N_EDGES + e];
    float m = __uint_as_float(mbits[dst]);     // non-negative: bits == float
    float v = __expf(ev[e] - m);
    ex[e] = v;
    atomicAdd(s + dst, v);
}

// ---------------------------------------------------------------------------
// vals = ex / s[dst]; scatter into dense sgraph[src, dst].
// Duplicate edges write identical values -> order-independent.
// ---------------------------------------------------------------------------
__global__ __launch_bounds__(256) void edge_scatter_kernel(const int* __restrict__ edge,
                                                           const float* __restrict__ ex,
                                                           const float* __restrict__ s,
                                                           float* __restrict__ sg) {
    int e = blockIdx.x * 256 + threadIdx.x;
    if (e >= N_EDGES) return;
    int src = edge[e];
    int dst = edge[N_EDGES + e];
    sg[(size_t)src * N_NODES + dst] = ex[e] / s[dst];
}

// ---------------------------------------------------------------------------
extern "C" void kernel_launch(void* const* d_in, const int* in_sizes, int n_in,
                              void* d_out, int out_size, void* d_ws, size_t ws_size,
                              hipStream_t stream) {
    const float* x    = (const float*)d_in[0];
    const float* w    = (const float*)d_in[1];
    const float* a    = (const float*)d_in[2];
    const int*   edge = (const int*)d_in[3];

    float* h  = (float*)d_out;
    float* sg = (float*)d_out + (size_t)N_NODES * DOUT;

    float*    ev    = (float*)d_ws;
    float*    ex    = ev + N_EDGES;
    unsigned* mbits = (unsigned*)(ex + N_EDGES);
    float*    s     = (float*)(mbits + N_NODES);

    // 1. zero segment max/sum
    zero_ms_kernel<<<N_NODES / 256, 256, 0, stream>>>(mbits, s);
    // 2. zero the dense 1.07 GB sgraph (dominant memory cost)
    zero_fill_kernel<<<8192, 256, 0, stream>>>((v4f*)sg,
                                               (long long)N_NODES * N_NODES / 4);
    // 3. h = x @ W via fp32 WMMA
    gemm_wmma_f32_kernel<<<1024, 128, 0, stream>>>(x, w, h);
    // 4. edge scores + segment max
    edge_score_kernel<<<N_EDGES / 8, 256, 0, stream>>>(h, a, edge, ev, mbits);
    // 5. exp + segment sum
    edge_exp_kernel<<<N_EDGES / 256, 256, 0, stream>>>(edge, ev, mbits, ex, s);
    // 6. normalize + scatter into dense output
    edge_scatter_kernel<<<N_EDGES / 256, 256, 0, stream>>>(edge, ex, s, sg);
}